// GraphEncoder_22058952032468
// MI455X (gfx1250) — compile-verified
//
#include <hip/hip_runtime.h>

typedef float v2f __attribute__((ext_vector_type(2)));
typedef float v8f __attribute__((ext_vector_type(8)));

#define HDIM 256

// ---------------------------------------------------------------------------
// Degree histogram: deg[dst[e]] += 1
// ---------------------------------------------------------------------------
__global__ void degree_kernel(const int* __restrict__ dst, float* __restrict__ deg, int E) {
    int e = blockIdx.x * blockDim.x + threadIdx.x;
    if (e < E) unsafeAtomicAdd(&deg[dst[e]], 1.0f);
}

// deg[i] -> 1 / max(deg[i], 1)
__global__ void rdenom_kernel(float* __restrict__ deg, int N) {
    int i = blockIdx.x * blockDim.x + threadIdx.x;
    if (i < N) deg[i] = 1.0f / fmaxf(deg[i], 1.0f);
}

// ---------------------------------------------------------------------------
// Node projection: h = x @ w_node + b_node   ([N,64] @ [64,256])
// 4 waves / block; each wave owns 4 column tiles (64 cols) of one 16-row slab.
// One A load feeds 4 V_WMMA_F32_16X16X4_F32 per K-step.
// ---------------------------------------------------------------------------
__global__ void __launch_bounds__(128)
node_proj_wmma(const float* __restrict__ x,      // [N,64]
               const float* __restrict__ w,      // [64,256]
               const float* __restrict__ bias,   // [256]
               float* __restrict__ h,            // [N,256]
               int N) {
    const int lane = threadIdx.x;                 // 0..31
    const int m0   = blockIdx.x * 16;
    const int n0   = threadIdx.y * 64;            // wave's 64-col strip
    int mrow = m0 + (lane & 15);
    if (mrow >= N) mrow = N - 1;                  // clamp (dead rows)
    const int nlane = lane & 15;
    const int khalf = (lane >> 4) * 2;            // 0 or 2

    v8f c0 = {}, c1 = {}, c2 = {}, c3 = {};
    const float* ap_base = x + (size_t)mrow * 64 + khalf;
    const float* bp_base = w + (size_t)khalf * HDIM + n0 + nlane;
#pragma unroll
    for (int k = 0; k < 64; k += 4) {
        v2f a;
        const float* ap = ap_base + k;
        a.x = ap[0];
        a.y = ap[1];
        const float* bp = bp_base + (size_t)k * HDIM;
        v2f b0, b1, b2, b3;
        b0.x = bp[0];   b0.y = bp[HDIM];
        b1.x = bp[16];  b1.y = bp[HDIM + 16];
        b2.x = bp[32];  b2.y = bp[HDIM + 32];
        b3.x = bp[48];  b3.y = bp[HDIM + 48];
        c0 = __builtin_amdgcn_wmma_f32_16x16x4_f32(false, a, false, b0, (short)0, c0, false, false);
        c1 = __builtin_amdgcn_wmma_f32_16x16x4_f32(false, a, false, b1, (short)0, c1, false, false);
        c2 = __builtin_amdgcn_wmma_f32_16x16x4_f32(false, a, false, b2, (short)0, c2, false, false);
        c3 = __builtin_amdgcn_wmma_f32_16x16x4_f32(false, a, false, b3, (short)0, c3, false, false);
    }

    const int mbase = m0 + ((lane >> 4) * 8);
    v8f acc[4] = {c0, c1, c2, c3};
#pragma unroll
    for (int t = 0; t < 4; ++t) {
        const int ncol = n0 + t * 16 + nlane;
        const float bv = bias[ncol];
#pragma unroll
        for (int i = 0; i < 8; ++i) {
            int m = mbase + i;
            if (m < N) h[(size_t)m * HDIM + ncol] = acc[t][i] + bv;
        }
    }
}

// ---------------------------------------------------------------------------
// Scatter: msg[dst[e]] += h[src[e]]   (64 lanes per edge, float4 per lane)
// h and msg both ~102 MB -> resident in the 192 MB L2; atomics RMW in L2.
// ---------------------------------------------------------------------------
__global__ void __launch_bounds__(256)
scatter_add(const float* __restrict__ h,
            const int* __restrict__ src,
            const int* __restrict__ dst,
            float* __restrict__ msg, int E) {
    long long gid = (long long)blockIdx.x * blockDim.x + threadIdx.x;
    int e  = (int)(gid >> 6);
    int c4 = ((int)gid & 63) * 4;
    if (e >= E) return;
    const int s = src[e];
    const int d = dst[e];
    const float4 v = *reinterpret_cast<const float4*>(h + (size_t)s * HDIM + c4);
    float* mp = msg + (size_t)d * HDIM + c4;
    unsafeAtomicAdd(mp + 0, v.x);
    unsafeAtomicAdd(mp + 1, v.y);
    unsafeAtomicAdd(mp + 2, v.z);
    unsafeAtomicAdd(mp + 3, v.w);
}

// ---------------------------------------------------------------------------
// GNN layer: h = relu(h + (msg * rdeg_row) @ w + b), in-place on h.
// [N,256]@[256,256] via f32 WMMA; per-row 1/deg folded into A loads.
// Same 4-tile-per-wave structure as node_proj.
// ---------------------------------------------------------------------------
__global__ void __launch_bounds__(128)
gnn_layer_wmma(const float* __restrict__ msg,    // [N,256]
               const float* __restrict__ rdeg,   // [N]
               const float* __restrict__ w,      // [256,256]
               const float* __restrict__ bias,   // [256]
               float* __restrict__ h,            // [N,256] in/out
               int N) {
    const int lane = threadIdx.x;
    const int m0   = blockIdx.x * 16;
    const int n0   = threadIdx.y * 64;
    int mrow = m0 + (lane & 15);
    if (mrow >= N) mrow = N - 1;
    const int nlane = lane & 15;
    const int khalf = (lane >> 4) * 2;
    const float scale = rdeg[mrow];

    v8f c0 = {}, c1 = {}, c2 = {}, c3 = {};
    const float* ap_base = msg + (size_t)mrow * HDIM + khalf;
    const float* bp_base = w + (size_t)khalf * HDIM + n0 + nlane;
#pragma unroll 4
    for (int k = 0; k < HDIM; k += 4) {
        v2f a;
        const float* ap = ap_base + k;
        a.x = ap[0] * scale;
        a.y = ap[1] * scale;
        const float* bp = bp_base + (size_t)k * HDIM;
        v2f b0, b1, b2, b3;
        b0.x = bp[0];   b0.y = bp[HDIM];
        b1.x = bp[16];  b1.y = bp[HDIM + 16];
        b2.x = bp[32];  b2.y = bp[HDIM + 32];
        b3.x = bp[48];  b3.y = bp[HDIM + 48];
        c0 = __builtin_amdgcn_wmma_f32_16x16x4_f32(false, a, false, b0, (short)0, c0, false, false);
        c1 = __builtin_amdgcn_wmma_f32_16x16x4_f32(false, a, false, b1, (short)0, c1, false, false);
        c2 = __builtin_amdgcn_wmma_f32_16x16x4_f32(false, a, false, b2, (short)0, c2, false, false);
        c3 = __builtin_amdgcn_wmma_f32_16x16x4_f32(false, a, false, b3, (short)0, c3, false, false);
    }

    const int mbase = m0 + ((lane >> 4) * 8);
    v8f acc[4] = {c0, c1, c2, c3};
#pragma unroll
    for (int t = 0; t < 4; ++t) {
        const int ncol = n0 + t * 16 + nlane;
        const float bv = bias[ncol];
#pragma unroll
        for (int i = 0; i < 8; ++i) {
            int m = mbase + i;
            if (m < N) {
                size_t idx = (size_t)m * HDIM + ncol;
                float v = acc[t][i] + h[idx] + bv;  // skip connection + bias
                h[idx] = fmaxf(v, 0.0f);            // relu, in-place (1 reader)
            }
        }
    }
}

// ---------------------------------------------------------------------------
// Column sum: g[c] += sum over a row-chunk of h[:, c]
// ---------------------------------------------------------------------------
__global__ void __launch_bounds__(256)
col_sum_kernel(const float* __restrict__ h, float* __restrict__ g, int N) {
    const int col = threadIdx.x;
    const int rows_per_block = (N + gridDim.x - 1) / gridDim.x;
    const int r0 = blockIdx.x * rows_per_block;
    const int r1 = min(r0 + rows_per_block, N);
    float acc = 0.0f;
    for (int r = r0; r < r1; ++r) acc += h[(size_t)r * HDIM + col];
    unsafeAtomicAdd(&g[col], acc);
}

// ---------------------------------------------------------------------------
// Head MLP: out = relu(g/N @ w1 + b1) @ w2 + b2   (single block, 256 thr)
// ---------------------------------------------------------------------------
__global__ void __launch_bounds__(256)
head_kernel(const float* __restrict__ g,
            const float* __restrict__ w1, const float* __restrict__ b1,
            const float* __restrict__ w2, const float* __restrict__ b2,
            float* __restrict__ out, float invN) {
    __shared__ float gs[HDIM];
    __shared__ float ts[HDIM];
    const int t = threadIdx.x;
    gs[t] = g[t] * invN;
    __syncthreads();
    float acc = b1[t];
#pragma unroll 8
    for (int i = 0; i < HDIM; ++i) acc = fmaf(gs[i], w1[(size_t)i * HDIM + t], acc);
    ts[t] = fmaxf(acc, 0.0f);
    __syncthreads();
    float acc2 = b2[t];
#pragma unroll 8
    for (int i = 0; i < HDIM; ++i) acc2 = fmaf(ts[i], w2[(size_t)i * HDIM + t], acc2);
    out[t] = acc2;
}

// ---------------------------------------------------------------------------
extern "C" void kernel_launch(void* const* d_in, const int* in_sizes, int n_in,
                              void* d_out, int out_size, void* d_ws, size_t ws_size,
                              hipStream_t stream) {
    const float* x      = (const float*)d_in[0];
    const int*   src    = (const int*)d_in[1];
    const int*   dst    = (const int*)d_in[2];
    const float* w_node = (const float*)d_in[3];
    const float* b_node = (const float*)d_in[4];
    const float* w_gnn  = (const float*)d_in[5];
    const float* b_gnn  = (const float*)d_in[6];
    const float* w_p1   = (const float*)d_in[7];
    const float* b_p1   = (const float*)d_in[8];
    const float* w_p2   = (const float*)d_in[9];
    const float* b_p2   = (const float*)d_in[10];
    float* out = (float*)d_out;

    const int N = in_sizes[0] / 64;   // D_NODE = 64
    const int E = in_sizes[1];

    // Workspace layout (carved from d_ws)
    char* ws = (char*)d_ws;
    size_t off = 0;
    auto carve = [&](size_t bytes) {
        void* p = ws + off;
        off += (bytes + 255) & ~(size_t)255;
        return p;
    };
    float* h   = (float*)carve((size_t)N * HDIM * sizeof(float));  // 102.4 MB
    float* msg = (float*)carve((size_t)N * HDIM * sizeof(float));  // 102.4 MB
    float* deg = (float*)carve((size_t)N * sizeof(float));
    float* g   = (float*)carve(HDIM * sizeof(float));
    (void)ws_size; (void)n_in; (void)out_size;

    hipMemsetAsync(deg, 0, (size_t)N * sizeof(float), stream);
    hipMemsetAsync(g,   0, HDIM * sizeof(float), stream);

    degree_kernel<<<(E + 255) / 256, 256, 0, stream>>>(dst, deg, E);
    rdenom_kernel<<<(N + 255) / 256, 256, 0, stream>>>(deg, N);

    const dim3 wblk(32, 4);                        // 4 waves; each owns 64 cols
    const dim3 wgrd((N + 15) / 16);                // one 16-row slab per block
    node_proj_wmma<<<wgrd, wblk, 0, stream>>>(x, w_node, b_node, h, N);

    for (int l = 0; l < 3; ++l) {
        hipMemsetAsync(msg, 0, (size_t)N * HDIM * sizeof(float), stream);
        long long work = (long long)E * 64;        // 64 lanes per edge
        scatter_add<<<(unsigned)((work + 255) / 256), 256, 0, stream>>>(h, src, dst, msg, E);
        gnn_layer_wmma<<<wgrd, wblk, 0, stream>>>(msg, deg,
                                                  w_gnn + (size_t)l * HDIM * HDIM,
                                                  b_gnn + (size_t)l * HDIM, h, N);
    }

    col_sum_kernel<<<256, 256, 0, stream>>>(h, g, N);
    head_kernel<<<1, 256, 0, stream>>>(g, w_p1, b_p1, w_p2, b_p2, out, 1.0f / (float)N);
}